// ContourletSD_32203664785750
// MI455X (gfx1250) — compile-verified
//
#include <hip/hip_runtime.h>
#include <hip/hip_bf16.h>

// ---------------------------------------------------------------------------
// Contourlet-SD for gfx1250: four-step DFT on f32 WMMA, TDM-staged operands.
// ---------------------------------------------------------------------------

typedef __attribute__((ext_vector_type(2))) float v2f;
typedef __attribute__((ext_vector_type(8))) float v8f;
typedef __attribute__((ext_vector_type(4))) unsigned int v4u;
typedef __attribute__((ext_vector_type(8))) int v8i;
typedef __attribute__((ext_vector_type(4))) int v4i;

#define PI_F 3.14159265358979323846f
#define LPB 32   // lines processed per block (amortizes LDS-staged operands)

__device__ __forceinline__ v8f wmma4(v2f a, v2f b, v8f c) {
  // V_WMMA_F32_16X16X4_F32 : D(16x16 f32) = A(16x4) * B(4x16) + C
  return __builtin_amdgcn_wmma_f32_16x16x4_f32(
      false, a, false, b, (short)0, c, false, false);
}

// Tensor Data Mover: 2-D tile (f32) global -> LDS. D# packed per ISA 8.3/8.4.
//   rowlen  : tile width in elements (contiguous)
//   nrows   : tile height
//   rstride : row stride in elements in global memory
__device__ __forceinline__ void tdm_load_2d(const float* gsrc, void* lds_dst,
                                            unsigned rowlen, unsigned nrows,
                                            unsigned rstride) {
  unsigned long long ga = (unsigned long long)(const void*)gsrc;
  unsigned lo = (unsigned)(unsigned long long)lds_dst;  // LDS byte offset = addr[31:0]
  v4u g0;
  g0[0] = 1u;                                           // count=1 (valid), user mode
  g0[1] = lo;                                           // lds_addr
  g0[2] = (unsigned)ga;                                 // global_addr[31:0]
  g0[3] = (unsigned)((ga >> 32) & 0x01ffffffu) | 0x80000000u; // addr[56:32] | type=2
  v8i g1;
  g1[0] = 0x20000;                                      // data_size=4B, mask=0
  g1[1] = (int)((rstride & 0xffffu) << 16);             // tensor_dim0[15:0]
  g1[2] = (int)((rstride >> 16) & 0xffffu)
        | (int)((nrows & 0xffffu) << 16);               // dim0[31:16] | dim1[15:0]
  g1[3] = (int)((nrows >> 16) & 0xffffu)
        | (int)((rowlen & 0xffffu) << 16);              // dim1[31:16] | tile_dim0
  g1[4] = (int)(nrows & 0xffffu);                       // tile_dim1 | tile_dim2=0
  g1[5] = (int)rstride;                                 // tensor_dim0_stride[31:0]
  g1[6] = 0;                                            // stride hi | dim1_stride lo
  g1[7] = 0;
  v4i z4 = {0, 0, 0, 0};
  v8i z8 = {0, 0, 0, 0, 0, 0, 0, 0};
  // 6-arg form on this toolchain: (g0, g1, g2, g3, g4, cpol)
  __builtin_amdgcn_tensor_load_to_lds(g0, g1, z4, z4, z8, 0);
}

__constant__ float PKVA_F[12] = {
    -0.0144f, 0.0272f, -0.0526f, 0.0972f, -0.193f, 0.63f,
     0.63f,  -0.193f,  0.0972f, -0.0526f, 0.0272f, -0.0144f };

// ------------------------------- tables ------------------------------------
// Layout (floats from start of ws):
//   d16c @0 (256)   d16sn @256 (-sin)  d16sp @512 (+sin)
//   d64c @768 (4096) d64sn @4864  d64sp @8960
//   d32c @13056 (1024) d32sn @14080 d32sp @15104
//   tw1024 @16128 : fc,fs,ic,is (1024 each)
//   tw512  @20224 : fc,fs,ic,is (512 each)
//   tw256  @22272 : fc,fs,ic,is (256 each)
//   phi1024 @23296, phi512 @24320, phi256 @24832   (tables end < 32768)

__device__ __forceinline__ float phi_val(int i, int N) {
  float f = (float)((i <= N / 2) ? i : N - i) / (float)N;
  float w = 2.f * PI_F * f;
  float s = (w - PI_F / 3.f) / (PI_F / 3.f);
  s = fminf(fmaxf(s, 0.f), 1.f);
  float b = s * s * s * s * (35.f - 84.f * s + 70.f * s * s - 20.f * s * s * s);
  return cosf(0.5f * PI_F * b);
}

__global__ void fill_tables(float* __restrict__ W) {
  long g = (long)blockIdx.x * blockDim.x + threadIdx.x;
  long gs = (long)gridDim.x * blockDim.x;
  for (long i = g; i < 256; i += gs) {               // D16
    int j = (int)(i >> 4), k = (int)(i & 15);
    float a = 2.f * PI_F * (float)(j * k) / 16.f;
    W[i] = cosf(a); W[256 + i] = -sinf(a); W[512 + i] = sinf(a);
  }
  for (long i = g; i < 4096; i += gs) {              // D64
    int j = (int)(i >> 6), k = (int)(i & 63);
    float a = 2.f * PI_F * (float)(j * k) / 64.f;
    W[768 + i] = cosf(a); W[4864 + i] = -sinf(a); W[8960 + i] = sinf(a);
  }
  for (long i = g; i < 1024; i += gs) {              // D32
    int j = (int)(i >> 5), k = (int)(i & 31);
    float a = 2.f * PI_F * (float)(j * k) / 32.f;
    W[13056 + i] = cosf(a); W[14080 + i] = -sinf(a); W[15104 + i] = sinf(a);
  }
  for (long i = g; i < 1024; i += gs) {              // tw1024 (M=64)
    int p = (int)(i >> 6), b = (int)(i & 63);
    float a = 2.f * PI_F * (float)(p * b) / 1024.f;
    float c = cosf(a), s = sinf(a);
    W[16128 + i] = c; W[17152 + i] = -s;
    W[18176 + i] = c * (1.f / 1024.f); W[19200 + i] = s * (1.f / 1024.f);
  }
  for (long i = g; i < 512; i += gs) {               // tw512 (M=32)
    int p = (int)(i >> 5), b = (int)(i & 31);
    float a = 2.f * PI_F * (float)(p * b) / 512.f;
    float c = cosf(a), s = sinf(a);
    W[20224 + i] = c; W[20736 + i] = -s;
    W[21248 + i] = c * (1.f / 512.f); W[21760 + i] = s * (1.f / 512.f);
  }
  for (long i = g; i < 256; i += gs) {               // tw256 (M=16)
    int p = (int)(i >> 4), b = (int)(i & 15);
    float a = 2.f * PI_F * (float)(p * b) / 256.f;
    float c = cosf(a), s = sinf(a);
    W[22272 + i] = c; W[22528 + i] = -s;
    W[22784 + i] = c * (1.f / 256.f); W[23040 + i] = s * (1.f / 256.f);
  }
  for (long i = g; i < 1024; i += gs) W[23296 + i] = phi_val((int)i, 1024);
  for (long i = g; i < 512;  i += gs) W[24320 + i] = phi_val((int)i, 512);
  for (long i = g; i < 256;  i += gs) W[24832 + i] = phi_val((int)i, 256);
}

// ---------------------- four-step DFT, WMMA step 1 -------------------------
// T'[p][b] = twiddle[p][b] * sum_a D16[p][a] * X[a*M + b]
// One wave per (b-tile, line-group); D16 planes + twiddle tile staged via TDM.
__global__ void dft_step1(const float* __restrict__ inR, const float* __restrict__ inI,
                          const float* __restrict__ a_c, const float* __restrict__ a_s,
                          const float* __restrict__ a_sn,
                          const float* __restrict__ twc, const float* __restrict__ tws,
                          float* __restrict__ tR, float* __restrict__ tI,
                          int M, int lpi, long lineStride, long imgStride, long estride) {
  __shared__ float sAc[256], sAs[256], sAn[256], sTc[256], sTs[256];
  const int  lane = threadIdx.x;
  const int  b0   = blockIdx.x * 16;
  // Stage stationary operands into LDS with the Tensor Data Mover.
  tdm_load_2d(a_c,      sAc, 256, 1, 256);
  tdm_load_2d(a_s,      sAs, 256, 1, 256);
  tdm_load_2d(a_sn,     sAn, 256, 1, 256);
  tdm_load_2d(twc + b0, sTc, 16, 16, (unsigned)M);
  tdm_load_2d(tws + b0, sTs, 16, 16, (unsigned)M);
  __builtin_amdgcn_s_wait_tensorcnt(0);   // single wave: no barrier needed

  const int m  = lane & 15;
  const int kb = (lane >> 4) << 1;
  const int col = b0 + m;
  const int rb = (lane >> 4) << 3;

  for (int l = 0; l < LPB; ++l) {
    const long line = (long)blockIdx.y * LPB + l;
    const long inBase = (line / lpi) * imgStride + (line % lpi) * lineStride;
    v8f accR = {}; v8f accI = {};
    for (int k4 = 0; k4 < 16; k4 += 4) {
      const int ka = k4 + kb;
      v2f aC, aS;
      aC.x = sAc[m * 16 + ka]; aC.y = sAc[m * 16 + ka + 1];
      aS.x = sAs[m * 16 + ka]; aS.y = sAs[m * 16 + ka + 1];
      const long e0 = inBase + (long)(ka * M + col) * estride;
      const long e1 = inBase + (long)((ka + 1) * M + col) * estride;
      v2f bR; bR.x = inR[e0]; bR.y = inR[e1];
      accR = wmma4(aC, bR, accR);
      accI = wmma4(aS, bR, accI);
      if (inI) {
        v2f aN; aN.x = sAn[m * 16 + ka]; aN.y = sAn[m * 16 + ka + 1];
        v2f bI; bI.x = inI[e0]; bI.y = inI[e1];
        accR = wmma4(aN, bI, accR);   // -As * Bi
        accI = wmma4(aC, bI, accI);   //  Ac * Bi
      }
    }
    const long outBase = line * (long)(M * 16);
#pragma unroll
    for (int i = 0; i < 8; ++i) {
      const int p = rb + i;
      const float tc = sTc[p * 16 + m];
      const float ts = sTs[p * 16 + m];
      const float r = accR[i], q = accI[i];
      tR[outBase + p * M + col] = r * tc - q * ts;
      tI[outBase + p * M + col] = r * ts + q * tc;
    }
  }
}

// ---------------------- four-step DFT, WMMA step 3 -------------------------
// Y[p + 16q] = sum_b T'[p][b] * DM[b][q]   (DM symmetric); DM tiles via TDM.
__global__ void dft_step3(const float* __restrict__ tR, const float* __restrict__ tI,
                          const float* __restrict__ dmc, const float* __restrict__ dms,
                          const float* __restrict__ dmsn,
                          float* __restrict__ outR, float* __restrict__ outI,
                          int M, int lpi, long lineStride, long imgStride, long estride) {
  __shared__ float sBc[1024], sBs[1024], sBn[1024];   // M x 16 column tiles
  const int  lane = threadIdx.x;
  const int  q0   = blockIdx.x * 16;
  tdm_load_2d(dmc  + q0, sBc, 16, (unsigned)M, (unsigned)M);
  tdm_load_2d(dms  + q0, sBs, 16, (unsigned)M, (unsigned)M);
  tdm_load_2d(dmsn + q0, sBn, 16, (unsigned)M, (unsigned)M);
  __builtin_amdgcn_s_wait_tensorcnt(0);

  const int m  = lane & 15;
  const int kb = (lane >> 4) << 1;
  const int rb = (lane >> 4) << 3;

  for (int l = 0; l < LPB; ++l) {
    const long line = (long)blockIdx.y * LPB + l;
    const long tBase = line * (long)(M * 16);
    v8f accR = {}; v8f accI = {};
    for (int k4 = 0; k4 < M; k4 += 4) {
      const int b = k4 + kb;
      __builtin_prefetch(tR + tBase + m * M + b + 16, 0, 1);
      v2f ar, ai, bc, bs, bn;
      ar.x = tR[tBase + m * M + b];  ar.y = tR[tBase + m * M + b + 1];
      ai.x = tI[tBase + m * M + b];  ai.y = tI[tBase + m * M + b + 1];
      bc.x = sBc[b * 16 + m];        bc.y = sBc[(b + 1) * 16 + m];
      bs.x = sBs[b * 16 + m];        bs.y = sBs[(b + 1) * 16 + m];
      bn.x = sBn[b * 16 + m];        bn.y = sBn[(b + 1) * 16 + m];
      accR = wmma4(ar, bc, accR);
      accR = wmma4(ai, bn, accR);
      accI = wmma4(ar, bs, accI);
      accI = wmma4(ai, bc, accI);
    }
    const long outBase = (line / lpi) * imgStride + (line % lpi) * lineStride;
#pragma unroll
    for (int i = 0; i < 8; ++i) {
      const int p = rb + i;
      const long u = (long)p + 16L * (q0 + m);
      outR[outBase + u * estride] = accR[i];
      if (outI) outI[outBase + u * estride] = accI[i];
    }
  }
}

// ------------------------------ pointwise ----------------------------------
__global__ void mask_mul(const float* __restrict__ xr, const float* __restrict__ xi,
                         const float* __restrict__ phi,
                         float* __restrict__ outR, float* __restrict__ outI,
                         int N, int lp, long total) {
  long g = (long)blockIdx.x * blockDim.x + threadIdx.x;
  long gs = (long)gridDim.x * blockDim.x;
  for (long idx = g; idx < total; idx += gs) {
    int j = (int)(idx % N);
    int i = (int)((idx / N) % N);
    float L = phi[i] * phi[j];
    float m = lp ? L : sqrtf(fmaxf(0.f, 1.f - L * L));
    outR[idx] = xr[idx] * m;
    outI[idx] = xi[idx] * m;
  }
}

__global__ void decimate2(const float* __restrict__ in, float* __restrict__ out,
                          int N, long total) {
  long g = (long)blockIdx.x * blockDim.x + threadIdx.x;
  long gs = (long)gridDim.x * blockDim.x;
  const int H = N >> 1;
  for (long idx = g; idx < total; idx += gs) {
    int w = (int)(idx % H);
    long t = idx / H;
    int h = (int)(t % H);
    long b = t / H;
    out[idx] = in[b * (long)N * N + (long)(2 * h) * N + 2 * w];
  }
}

// --------------------------- DFB ladder stages -----------------------------
__global__ void fan_pre(const float* __restrict__ x, float* __restrict__ p0,
                        float* __restrict__ p1, int H, int W, int axisW, long total) {
  long g = (long)blockIdx.x * blockDim.x + threadIdx.x;
  long gs = (long)gridDim.x * blockDim.x;
  for (long idx = g; idx < total; idx += gs) {
    if (axisW) {
      const int W2 = W >> 1;
      int w2 = (int)(idx % W2);
      long t = idx / W2;
      int h = (int)(t % H);
      long s = t / H;
      int par = h & 1;
      float sg = par ? -1.f : 1.f;
      const float* row = x + s * (long)H * W + (long)h * W;
      int i0 = 2 * w2 + par;
      int i1 = 2 * w2 + 1 + par; if (i1 >= W) i1 -= W;
      p0[idx] = sg * row[i0];
      p1[idx] = sg * row[i1];
    } else {
      const int H2 = H >> 1;
      int w = (int)(idx % W);
      long t = idx / W;
      int h2 = (int)(t % H2);
      long s = t / H2;
      int par = w & 1;
      float sg = par ? -1.f : 1.f;
      const float* img = x + s * (long)H * W;
      int i0 = 2 * h2 + par;
      int i1 = 2 * h2 + 1 + par; if (i1 >= H) i1 -= H;
      p0[idx] = sg * img[(long)i0 * W + w];
      p1[idx] = sg * img[(long)i1 * W + w];
    }
  }
}

__global__ void fan_ladder(const float* __restrict__ src, const float* __restrict__ base,
                           float* __restrict__ out, float alpha,
                           int Hc, int Wc, int axisW, long total) {
  long g = (long)blockIdx.x * blockDim.x + threadIdx.x;
  long gs = (long)gridDim.x * blockDim.x;
  for (long idx = g; idx < total; idx += gs) {
    float acc = 0.f;
    if (axisW) {
      int w = (int)(idx % Wc);
      long rowBase = idx - w;
#pragma unroll
      for (int k = 0; k < 12; ++k) {
        int j = w - 6 + k;
        if (j < 0) j += Wc; else if (j >= Wc) j -= Wc;
        acc += PKVA_F[k] * src[rowBase + j];
      }
    } else {
      int w = (int)(idx % Wc);
      long t = idx / Wc;
      int h = (int)(t % Hc);
      long colBase = (t - h) * Wc + w;
#pragma unroll
      for (int k = 0; k < 12; ++k) {
        int j = h - 6 + k;
        if (j < 0) j += Hc; else if (j >= Hc) j -= Hc;
        acc += PKVA_F[k] * src[colBase + (long)j * Wc];
      }
    }
    out[idx] = base[idx] + alpha * acc;
  }
}

// ------------------------------- host side ---------------------------------
struct DftTabs {
  const float *d16c, *d16s_f, *d16s_i;   // cos, -sin, +sin (16x16)
  const float *dmc, *dms_f, *dms_i;      // cos, -sin, +sin (MxM)
  const float *twfc, *twfs, *twic, *twis;
};

static void fft_pass(hipStream_t st, float* tR, float* tI,
                     const float* inR, const float* inI,
                     float* outR, float* outI,
                     int N, int M, const DftTabs& T, bool inv,
                     int lpi, long lineStride, long imgStride, long estride,
                     int lpiO, long lineStrideO, long imgStrideO, long estrideO) {
  dim3 blk(32);
  dim3 g((unsigned)(M / 16), (unsigned)((16 * N) / LPB));
  const float* as  = inv ? T.d16s_i : T.d16s_f;
  const float* asn = inv ? T.d16s_f : T.d16s_i;
  const float* twc = inv ? T.twic : T.twfc;
  const float* tws = inv ? T.twis : T.twfs;
  dft_step1<<<g, blk, 0, st>>>(inR, inI, T.d16c, as, asn, twc, tws,
                               tR, tI, M, lpi, lineStride, imgStride, estride);
  const float* bs  = inv ? T.dms_i : T.dms_f;
  const float* bsn = inv ? T.dms_f : T.dms_i;
  dft_step3<<<g, blk, 0, st>>>(tR, tI, T.dmc, bs, bsn, outR, outI,
                               M, lpiO, lineStrideO, imgStrideO, estrideO);
}

extern "C" void kernel_launch(void* const* d_in, const int* in_sizes, int n_in,
                              void* d_out, int out_size, void* d_ws, size_t ws_size,
                              hipStream_t stream) {
  (void)in_sizes; (void)n_in; (void)out_size; (void)ws_size;

  float* Wt = (float*)d_ws;
  fill_tables<<<dim3(64), dim3(256), 0, stream>>>(Wt);

  const long SL = 16777216;                 // one 16x1024x1024 f32 plane
  float* CR0 = Wt + 32768;  float* CI0 = CR0 + SL;
  float* CR1 = CI0 + SL;    float* CI1 = CR1 + SL;
  float* CR2 = CI1 + SL;    float* CI2 = CR2 + SL;
  float* RB  = CI2 + SL;    float* RW  = RB + SL;     // real band / real LP
  float* X1  = RW + SL;                              // 16x512x512
  float* X2  = X1 + 4194304;                         // 16x256x256

  float* out = (float*)d_out;
  float* y0 = out;                  // (16,1,128,128)
  float* o1 = out + 262144;         // nlev=2  (4,16,1,128,128)
  float* o2 = out + 1310720;        // nlev=3  (8,16,1,256,128)
  float* o3 = out + 5505024;        // nlev=4  (16,16,1,256,256)

  const float* x0 = (const float*)d_in[0];

  for (int lev = 0; lev < 3; ++lev) {
    const int N = (lev == 0) ? 1024 : (lev == 1 ? 512 : 256);
    const int M = N / 16;
    DftTabs T;
    T.d16c = Wt + 0; T.d16s_f = Wt + 256; T.d16s_i = Wt + 512;
    if (M == 64)      { T.dmc = Wt + 768;   T.dms_f = Wt + 4864;  T.dms_i = Wt + 8960; }
    else if (M == 32) { T.dmc = Wt + 13056; T.dms_f = Wt + 14080; T.dms_i = Wt + 15104; }
    else              { T.dmc = Wt + 0;     T.dms_f = Wt + 256;   T.dms_i = Wt + 512; }
    const float* twb = (lev == 0) ? Wt + 16128 : (lev == 1 ? Wt + 20224 : Wt + 22272);
    T.twfc = twb; T.twfs = twb + N; T.twic = twb + 2 * N; T.twis = twb + 3 * N;
    const float* phi = (lev == 0) ? Wt + 23296 : (lev == 1 ? Wt + 24320 : Wt + 24832);

    const float* xin = (lev == 0) ? x0 : (lev == 1 ? X1 : X2);
    float* xnext = (lev == 0) ? X1 : (lev == 1 ? X2 : y0);
    float* dst   = (lev == 0) ? o3 : (lev == 1 ? o2 : o1);
    const int nlev = (lev == 0) ? 4 : (lev == 1 ? 3 : 2);
    const long NN = (long)N * N, tot = 16 * NN;

    // forward fft2: rows then cols -> (CR1,CI1)
    fft_pass(stream, CR0, CI0, xin, nullptr, CR1, CI1, N, M, T, false,
             N, N, NN, 1,  N, N, NN, 1);
    fft_pass(stream, CR0, CI0, CR1, CI1, CR1, CI1, N, M, T, false,
             N, 1, NN, N,  N, 1, NN, N);
    // high-pass band -> RB (real)
    mask_mul<<<dim3(2048), 256, 0, stream>>>(CR1, CI1, phi, CR2, CI2, N, 0, tot);
    fft_pass(stream, CR0, CI0, CR2, CI2, CR2, CI2, N, M, T, true,
             N, N, NN, 1,  N, N, NN, 1);
    fft_pass(stream, CR0, CI0, CR2, CI2, RB, nullptr, N, M, T, true,
             N, 1, NN, N,  N, 1, NN, N);
    // low-pass -> RW (real) -> decimate
    mask_mul<<<dim3(2048), 256, 0, stream>>>(CR1, CI1, phi, CR2, CI2, N, 1, tot);
    fft_pass(stream, CR0, CI0, CR2, CI2, CR2, CI2, N, M, T, true,
             N, N, NN, 1,  N, N, NN, 1);
    fft_pass(stream, CR0, CI0, CR2, CI2, RW, nullptr, N, M, T, true,
             N, 1, NN, N,  N, 1, NN, N);
    decimate2<<<dim3(2048), 256, 0, stream>>>(RW, xnext, N, tot / 4);

    // directional filter bank (ladder), ping-pong RB <-> RW, p0/p1 in CR0.
    float* p0b = CR0;
    float* p1b = CR0 + (SL >> 1);
    float* bufs[2] = { RB, RW };
    int cur = 0, H = N, Wd = N, axisW = 1;
    for (int st2 = 0; st2 < nlev; ++st2) {
      const long half = tot / 2;
      float* IN  = bufs[cur];
      float* OUT = (st2 == nlev - 1) ? dst : bufs[1 - cur];
      fan_pre<<<dim3(2048), 256, 0, stream>>>(IN, p0b, p1b, H, Wd, axisW, half);
      const int Hc = axisW ? H : (H >> 1);
      const int Wc = axisW ? (Wd >> 1) : Wd;
      float* D = OUT + half;   // d-subbands: second half of concat order
      float* S = OUT;          // s-subbands: first half
      fan_ladder<<<dim3(2048), 256, 0, stream>>>(p0b, p1b, D, -1.0f, Hc, Wc, axisW, half);
      fan_ladder<<<dim3(2048), 256, 0, stream>>>(D, p0b, S, 0.5f, Hc, Wc, axisW, half);
      if (axisW) Wd >>= 1; else H >>= 1;
      axisW ^= 1; cur ^= 1;
    }
  }
}